// BahdanauAttention_841813590086
// MI455X (gfx1250) — compile-verified
//
#include <hip/hip_runtime.h>
#include <hip/hip_bf16.h>
#include <math.h>

// ---- sizes (match reference) ----
#define B_  8
#define QL_ 128
#define ML_ 1024
#define QS_ 512
#define KS_ 512
#define H_  256

typedef float v2f __attribute__((ext_vector_type(2)));
typedef float v4f __attribute__((ext_vector_type(4)));
typedef float v8f __attribute__((ext_vector_type(8)));

// Full-precision fp32 WMMA: D(16x16) = A(16x4) * B(4x16) + C
__device__ __forceinline__ v8f wmma_f32_k4(v2f a, v2f b, v8f c) {
    // 8 args: (neg_a, A, neg_b, B, c_mod, C, reuse_a, reuse_b)
    return __builtin_amdgcn_wmma_f32_16x16x4_f32(false, a, false, b, (short)0, c,
                                                 false, false);
}

__device__ __forceinline__ float my_tanh(float x) {
#if __has_builtin(__builtin_amdgcn_tanhf)
    return __builtin_amdgcn_tanhf(x);   // V_TANH_F32 (gfx1250 TRANS op)
#else
    x = fminf(15.f, fmaxf(-15.f, x));
    float e = __expf(2.0f * x);         // v_exp_f32
    return (e - 1.0f) / (e + 1.0f);
#endif
}

// ---------------------------------------------------------------------------
// Kernel 1/2: OUT[M,N] = sum_k A[M,k] * W[N,k] + bias[N]
// A: Mrows x K row-major, W: N x K row-major (A * W^T), OUT: Mrows x N row-major.
// One wave per 16x16 output tile; 8 waves (tiles) per 256-thread block.
// A-operand layout (32-bit 16x4): lane L holds row M=L%16, K = v + 2*(L/16)
// -> contiguous v2f load. B-operand (4x16): lane L holds col N=L%16, same K rule.
// ---------------------------------------------------------------------------
__global__ __launch_bounds__(256) void proj_wmma_kernel(
    const float* __restrict__ A, const float* __restrict__ W,
    const float* __restrict__ bias, float* __restrict__ out,
    int K, int N, int ntiles) {

    const int lane = threadIdx.x & 31;
    const int wave = threadIdx.x >> 5;
    const int tile = blockIdx.x * 8 + wave;
    const int mt = tile / ntiles;
    const int nt = tile % ntiles;
    const int half = lane >> 4;   // 0 or 1
    const int l16  = lane & 15;

    const float* __restrict__ Arow = A + (size_t)(mt * 16 + l16) * K;
    const float* __restrict__ Wrow = W + (size_t)(nt * 16 + l16) * K;

    v8f acc = {};
    for (int k = 0; k < K; k += 4) {
        v2f a = *(const v2f*)(Arow + k + 2 * half);
        v2f b = *(const v2f*)(Wrow + k + 2 * half);
        acc = wmma_f32_k4(a, b, acc);
    }

    const float bv = bias[nt * 16 + l16];
    #pragma unroll
    for (int v = 0; v < 8; ++v) {
        int m = mt * 16 + v + 8 * half;           // C/D layout: M = v + 8*(lane/16)
        out[(size_t)m * N + nt * 16 + l16] = acc[v] + bv;
    }
}

// ---------------------------------------------------------------------------
// Kernel 3: logits + mask + softmax -> weights [B, QL, ML]
// One 256-thread workgroup per (b, q) row. Lane owns the 8 contiguous h values
// h = lane*8 .. lane*8+7 (two b128 loads per m-row, wave reads contiguous 1KB).
// Each wave handles m = wave, wave+8, ... ; shuffle-xor reduce 32 lanes per m
// (reduction over h is order-independent, so the blocked mapping is free).
// ---------------------------------------------------------------------------
__global__ __launch_bounds__(256) void logits_softmax_kernel(
    const float* __restrict__ qproj,   // [B*QL, H]
    const float* __restrict__ kproj,   // [B*ML, H]
    const int*   __restrict__ mask,    // [B, ML] (nonzero => masked out)
    const float* __restrict__ wl,      // [H]
    const float* __restrict__ blp,     // [1]
    float* __restrict__ weights) {     // [B*QL, ML]

    const int bq   = blockIdx.x;       // b*QL + q
    const int b    = bq >> 7;          // / QL_
    const int tid  = threadIdx.x;
    const int lane = tid & 31;
    const int wave = tid >> 5;

    __shared__ float lds_logit[ML_];
    __shared__ float red[8];

    const float* __restrict__ qrow  = qproj + (size_t)bq * H_;
    const float* __restrict__ kbase = kproj + (size_t)b * ML_ * H_;
    const int hbase = lane * 8;

    // Preload q-row chunk and wl chunk as vectors (32B aligned).
    v4f qh0 = *(const v4f*)(qrow + hbase);
    v4f qh1 = *(const v4f*)(qrow + hbase + 4);
    v4f wv0 = *(const v4f*)(wl + hbase);
    v4f wv1 = *(const v4f*)(wl + hbase + 4);
    const float bl = blp[0];

    for (int m = wave; m < ML_; m += 8) {
        const float* __restrict__ krow = kbase + (size_t)m * H_ + hbase;
        v4f k0 = *(const v4f*)(krow);
        v4f k1 = *(const v4f*)(krow + 4);
        float p = 0.f;
        #pragma unroll
        for (int j = 0; j < 4; ++j)
            p += wv0[j] * my_tanh(qh0[j] + k0[j]);
        #pragma unroll
        for (int j = 0; j < 4; ++j)
            p += wv1[j] * my_tanh(qh1[j] + k1[j]);
        #pragma unroll
        for (int off = 16; off >= 1; off >>= 1)
            p += __shfl_xor(p, off, 32);
        if (lane == 0) {
            float lg = p + bl;
            if (mask[b * ML_ + m] != 0) lg = -1e18f;
            lds_logit[m] = lg;
        }
    }
    __syncthreads();

    // --- max over ML ---
    float mx = -INFINITY;
    #pragma unroll
    for (int i = 0; i < 4; ++i)
        mx = fmaxf(mx, lds_logit[tid + 256 * i]);
    #pragma unroll
    for (int off = 16; off >= 1; off >>= 1)
        mx = fmaxf(mx, __shfl_xor(mx, off, 32));
    if (lane == 0) red[wave] = mx;
    __syncthreads();
    mx = red[0];
    #pragma unroll
    for (int w = 1; w < 8; ++w) mx = fmaxf(mx, red[w]);
    __syncthreads();

    // --- sum of exp ---
    float ex[4];
    float sm = 0.f;
    #pragma unroll
    for (int i = 0; i < 4; ++i) {
        ex[i] = __expf(lds_logit[tid + 256 * i] - mx);
        sm += ex[i];
    }
    #pragma unroll
    for (int off = 16; off >= 1; off >>= 1)
        sm += __shfl_xor(sm, off, 32);
    if (lane == 0) red[wave] = sm;
    __syncthreads();
    sm = 0.f;
    #pragma unroll
    for (int w = 0; w < 8; ++w) sm += red[w];
    const float inv = 1.0f / sm;

    float* __restrict__ wout = weights + (size_t)bq * ML_;
    #pragma unroll
    for (int i = 0; i < 4; ++i)
        wout[tid + 256 * i] = ex[i] * inv;
}

// ---------------------------------------------------------------------------
// Kernel 4: attns[b] = weights[b] (QLxML) @ memory[b] (MLxKS), fp32 WMMA.
// One wave per 16x16 output tile; 8 tiles per 256-thread block.
// ---------------------------------------------------------------------------
__global__ __launch_bounds__(256) void attn_gemm_kernel(
    const float* __restrict__ Wt,      // [B, QL, ML]
    const float* __restrict__ Mem,     // [B, ML, KS]
    float* __restrict__ Out) {         // [B, QL, KS]

    const int NT = KS_ / 16;           // 32
    const int MT = QL_ / 16;           // 8
    const int tilesPerB = NT * MT;     // 256

    const int lane = threadIdx.x & 31;
    const int wave = threadIdx.x >> 5;
    const int tile = blockIdx.x * 8 + wave;
    const int b  = tile / tilesPerB;
    const int r  = tile % tilesPerB;
    const int mt = r / NT;
    const int nt = r % NT;
    const int half = lane >> 4;
    const int l16  = lane & 15;

    const float* __restrict__ A  = Wt  + (size_t)b * QL_ * ML_;
    const float* __restrict__ Bm = Mem + (size_t)b * ML_ * KS_;
    float* __restrict__ O        = Out + (size_t)b * QL_ * KS_;

    const float* __restrict__ Arow = A + (size_t)(mt * 16 + l16) * ML_;
    const int ncol = nt * 16 + l16;

    v8f acc = {};
    for (int k = 0; k < ML_; k += 4) {
        v2f a = *(const v2f*)(Arow + k + 2 * half);
        v2f bb;
        bb.x = Bm[(size_t)(k + 2 * half)     * KS_ + ncol];
        bb.y = Bm[(size_t)(k + 2 * half + 1) * KS_ + ncol];
        acc = wmma_f32_k4(a, bb, acc);
    }

    #pragma unroll
    for (int v = 0; v < 8; ++v) {
        int m = mt * 16 + v + 8 * half;
        O[(size_t)m * KS_ + ncol] = acc[v];
    }
}

// ---------------------------------------------------------------------------
extern "C" void kernel_launch(void* const* d_in, const int* in_sizes, int n_in,
                              void* d_out, int out_size, void* d_ws, size_t ws_size,
                              hipStream_t stream) {
    const float* query  = (const float*)d_in[0];   // [B, QL, QS]
    const float* memory = (const float*)d_in[1];   // [B, ML, KS]
    const int*   mask   = (const int*)  d_in[2];   // [B, ML]
    const float* Wq     = (const float*)d_in[3];   // [H, QS]
    const float* bq     = (const float*)d_in[4];   // [H]
    const float* Wk     = (const float*)d_in[5];   // [H, KS]
    const float* bk     = (const float*)d_in[6];   // [H]
    const float* wl     = (const float*)d_in[7];   // [H]
    const float* bl     = (const float*)d_in[8];   // [1]

    float* attns   = (float*)d_out;                        // B*QL*QS
    float* weights = attns + (size_t)B_ * QL_ * QS_;       // B*QL*ML

    float* qproj = (float*)d_ws;                           // [B*QL, H]  1 MB
    float* kproj = qproj + (size_t)B_ * QL_ * H_;          // [B*ML, H]  8 MB

    const int ntilesH = H_ / 16;                           // 16

    // q projection: (B*QL/16)*(H/16) = 1024 tiles, 8 tiles/block
    {
        int tiles = (B_ * QL_ / 16) * ntilesH;
        proj_wmma_kernel<<<tiles / 8, 256, 0, stream>>>(query, Wq, bq, qproj,
                                                        QS_, H_, ntilesH);
    }
    // k projection: (B*ML/16)*(H/16) = 8192 tiles
    {
        int tiles = (B_ * ML_ / 16) * ntilesH;
        proj_wmma_kernel<<<tiles / 8, 256, 0, stream>>>(memory, Wk, bk, kproj,
                                                        KS_, H_, ntilesH);
    }
    // logits + masked softmax -> weights (also part of d_out)
    logits_softmax_kernel<<<B_ * QL_, 256, 0, stream>>>(qproj, kproj, mask,
                                                        wl, bl, weights);
    // attns = weights @ memory (fp32 WMMA), tiles = B*(QL/16)*(KS/16) = 2048
    {
        int tiles = B_ * (QL_ / 16) * (KS_ / 16);
        attn_gemm_kernel<<<tiles / 8, 256, 0, stream>>>(weights, memory, attns);
    }
}